// KnnGraph_51548197487015
// MI455X (gfx1250) — compile-verified
//
#include <hip/hip_runtime.h>

#define NN      8192
#define K1      31                 // K+1 = 31 neighbors kept per row/column
#define NSLICE  32
#define RPS     (NN / NSLICE)      // 256 rows per slice
#define TPB     256
#define NCAND   (NSLICE * K1)      // 992 candidates per column at merge

typedef float    f32x4 __attribute__((ext_vector_type(4)));
typedef unsigned u32x4 __attribute__((ext_vector_type(4)));

// ---------- order-preserving float <-> uint maps (ascending) ----------
__device__ __forceinline__ unsigned f2u_bits(unsigned u) {
    return (u & 0x80000000u) ? ~u : (u | 0x80000000u);
}
__device__ __forceinline__ float u2f(unsigned u) {
    unsigned v = (u & 0x80000000u) ? (u & 0x7fffffffu) : ~u;
    return __uint_as_float(v);
}

// ---------- register-resident sorted top-K1 insert (top[0] = largest) ----------
__device__ __forceinline__ void topk_insert(float (&top)[K1], float v) {
    float x = v;
#pragma unroll
    for (int i = 0; i < K1; ++i) {
        float hi = fmaxf(top[i], x);
        float lo = fminf(top[i], x);
        top[i] = hi;
        x = lo;
    }
}

// ---------- block-cooperative radix select: key of k-th largest ----------
// keys: LDS array of monotone uint keys, len elements; 256 threads.
__device__ __forceinline__ unsigned radix_select_kth(
    const unsigned* __restrict__ keys, int len, unsigned k,
    unsigned* hist, unsigned* sscan, unsigned* sel, int tid)
{
    unsigned pattern = 0u, pmask = 0u;
    for (int pass = 0; pass < 4; ++pass) {
        const int shift = 24 - 8 * pass;
        hist[tid] = 0u;
        __syncthreads();
        for (int i = tid; i < len; i += TPB) {
            unsigned u = keys[i];
            if ((u & pmask) == pattern)
                atomicAdd(&hist[(u >> shift) & 0xffu], 1u);
        }
        __syncthreads();
        // suffix-inclusive scan over 256 bins (Hillis-Steele)
        sscan[tid] = hist[tid];
        __syncthreads();
        for (int d = 1; d < 256; d <<= 1) {
            unsigned add = (tid + d < 256) ? sscan[tid + d] : 0u;
            __syncthreads();
            sscan[tid] += add;
            __syncthreads();
        }
        unsigned ge = sscan[tid];
        unsigned gt = (tid + 1 < 256) ? sscan[tid + 1] : 0u;
        if (ge >= k && gt < k) { sel[0] = (unsigned)tid; sel[1] = k - gt; }
        __syncthreads();
        pattern |= (sel[0] << shift);
        pmask   |= (0xffu << shift);
        k = sel[1];
        __syncthreads();
    }
    return pattern;   // exact key of the k-th largest
}

// ---------- kernel 1: per-(column, row-slice) partial top-31 lists ----------
__global__ __launch_bounds__(TPB) void col_partial_kernel(
    const float* __restrict__ a, float* __restrict__ partial)
{
    const int c = blockIdx.x * TPB + threadIdx.x;   // column
    const int s = blockIdx.y;                       // row slice
    const float* __restrict__ p = a + (size_t)(s * RPS) * NN + c;

    float top[K1];
#pragma unroll
    for (int i = 0; i < K1; ++i) top[i] = -3.402823466e38f;
    float vmin = top[K1 - 1];

    for (int r = 0; r < RPS; r += 8) {
        float v[8];
#pragma unroll
        for (int u = 0; u < 8; ++u) v[u] = p[(size_t)(r + u) * NN];
#pragma unroll
        for (int u = 0; u < 8; ++u) {
            if (v[u] > vmin) { topk_insert(top, v[u]); vmin = top[K1 - 1]; }
        }
    }
    // transposed layout -> coalesced stores: partial[(s*K1 + j)*NN + c]
#pragma unroll
    for (int j = 0; j < K1; ++j)
        partial[(size_t)(s * K1 + j) * NN + c] = top[j];
}

// ---------- kernel 2: merge 32 partial lists per column -> Tcol ----------
__global__ __launch_bounds__(TPB) void col_merge_kernel(
    const float* __restrict__ partial, float* __restrict__ tcol)
{
    __shared__ unsigned keys[NCAND];
    __shared__ unsigned hist[256];
    __shared__ unsigned sscan[256];
    __shared__ unsigned sel[2];

    const int tid = threadIdx.x;
    const int c   = blockIdx.x;

    for (int e = tid; e < NCAND; e += TPB) {
        unsigned u = __float_as_uint(partial[(size_t)e * NN + c]);
        keys[e] = f2u_bits(u);
    }
    __syncthreads();
    unsigned kth = radix_select_kth(keys, NCAND, K1, hist, sscan, sel, tid);
    if (tid == 0) tcol[c] = u2f(kth);
}

// ---------- kernel 3: row threshold + fused masked output ----------
__global__ __launch_bounds__(TPB) void row_select_apply_kernel(
    const float* __restrict__ a, const float* __restrict__ tcol,
    float* __restrict__ out)
{
    __shared__ unsigned keys[NN];       // 32 KB: one full row as monotone keys
    __shared__ unsigned hist[256];
    __shared__ unsigned sscan[256];
    __shared__ unsigned sel[2];

    const int tid = threadIdx.x;
    const int row = blockIdx.x;
    const float* __restrict__ grow = a + (size_t)row * NN;

    // ---- stage row into LDS with CDNA5 async global->LDS copies ----
    {
        unsigned           lds_base = (unsigned)(unsigned long long)&keys[0];
        unsigned long long gbase    = (unsigned long long)grow;
#pragma unroll
        for (int j = 0; j < 8; ++j) {
            unsigned off = ((unsigned)tid + (unsigned)j * TPB) * 16u;  // 16B/lane
            unsigned           laddr = lds_base + off;
            unsigned long long gaddr = gbase + off;
            asm volatile("global_load_async_to_lds_b128 %0, %1, off"
                         :: "v"(laddr), "v"(gaddr) : "memory");
        }
        asm volatile("s_wait_asynccnt 0x0" ::: "memory");
    }
    __syncthreads();

    // convert raw float bits -> monotone keys, in place
#pragma unroll
    for (int j = 0; j < NN / TPB; ++j) {
        int idx = tid + j * TPB;
        keys[idx] = f2u_bits(keys[idx]);
    }
    __syncthreads();

    const unsigned kth  = radix_select_kth(keys, NN, K1, hist, sscan, sel, tid);
    const float    trow = u2f(kth);

    // masked write-back: out = a * (a>=Trow) * (a>=Tcol) * (i!=j), streamed NT
#pragma unroll
    for (int j = 0; j < NN / (TPB * 4); ++j) {
        int c4 = tid * 4 + j * (TPB * 4);
        u32x4 kk = *(const u32x4*)&keys[c4];
        f32x4 tc = *(const f32x4*)(tcol + c4);
        float v0 = u2f(kk.x), v1 = u2f(kk.y), v2 = u2f(kk.z), v3 = u2f(kk.w);
        f32x4 o;
        o.x = (v0 >= trow && v0 >= tc.x && (c4 + 0) != row) ? v0 : 0.0f;
        o.y = (v1 >= trow && v1 >= tc.y && (c4 + 1) != row) ? v1 : 0.0f;
        o.z = (v2 >= trow && v2 >= tc.z && (c4 + 2) != row) ? v2 : 0.0f;
        o.w = (v3 >= trow && v3 >= tc.w && (c4 + 3) != row) ? v3 : 0.0f;
        __builtin_nontemporal_store(o, (f32x4*)(out + (size_t)row * NN + c4));
    }
}

extern "C" void kernel_launch(void* const* d_in, const int* in_sizes, int n_in,
                              void* d_out, int out_size, void* d_ws, size_t ws_size,
                              hipStream_t stream) {
    (void)in_sizes; (void)n_in; (void)out_size; (void)ws_size;
    const float* a    = (const float*)d_in[0];
    float*       out  = (float*)d_out;
    float*       tcol = (float*)d_ws;                           // 8192 floats
    float*       part = (float*)((char*)d_ws + 32768);          // 32*31*8192 floats (~31 MB)

    col_partial_kernel<<<dim3(NN / TPB, NSLICE), TPB, 0, stream>>>(a, part);
    col_merge_kernel<<<NN, TPB, 0, stream>>>(part, tcol);
    row_select_apply_kernel<<<NN, TPB, 0, stream>>>(a, tcol, out);
}